// OPAP_66219805770053
// MI455X (gfx1250) — compile-verified
//
#include <hip/hip_runtime.h>

typedef float v2f __attribute__((ext_vector_type(2)));
typedef float v8f __attribute__((ext_vector_type(8)));

#define NONE_ID (-2147483647 - 1)
#define BIG_ID  0x7fffffff

// ---------------------------------------------------------------------------
// Kernel 1: per-sample segment scan (general reference semantics), wave32
// shuffle-scan version (3 block barriers per 256-element chunk).
// Computes seg[b,t] (compacted word index, T for invalid), seg_start[b,s],
// counts[b,s], nseg[b], rowlim[b] (= last valid position + 1).
// ---------------------------------------------------------------------------
__global__ void opap_scan_kernel(const int* __restrict__ word_ids,
                                 int* __restrict__ seg,
                                 int* __restrict__ seg_start,
                                 int* __restrict__ counts,
                                 int* __restrict__ nseg_arr,
                                 int* __restrict__ rowlim_arr,
                                 int T) {
  const int b = blockIdx.x;
  const int tid = threadIdx.x;
  const int lane = tid & 31;
  const int wave = tid >> 5;
  __shared__ int s_wval[8];   // per-wave "last valid id" totals
  __shared__ int s_wsum[8];   // per-wave new-segment flag sums
  __shared__ int sh_last, sh_seg, sh_row;
  if (tid == 0) { sh_last = NONE_ID; sh_seg = 0; sh_row = 0; }
  __syncthreads();

  const int nch = (T + 255) >> 8;
  for (int c = 0; c < nch; ++c) {
    const int t = (c << 8) + tid;
    const int id = (t < T) ? word_ids[b * T + t] : -1;
    const bool valid = (t < T) && (id >= 0);
    const int carryLast = sh_last;
    const int carrySeg  = sh_seg;

    // wave scan 1: most recent valid word id at-or-before lane (in-register)
    int v = valid ? id : NONE_ID;
#pragma unroll
    for (int off = 1; off < 32; off <<= 1) {
      int o = __shfl_up(v, off, 32);
      if (lane >= off && v == NONE_ID) v = o;
    }
    if (lane == 31) s_wval[wave] = v;
    __syncthreads();                                   // B1

    // carry into this wave = last non-NONE of {chunk carry, earlier waves}
    int carry_w = carryLast;
    for (int wi = wave - 1; wi >= 0; --wi) {
      int x = s_wval[wi];
      if (x != NONE_ID) { carry_w = x; break; }
    }
    int p = __shfl_up(v, 1, 32);                       // exclusive last-valid
    if (lane == 0 || p == NONE_ID) p = carry_w;
    const int f = (valid && (p == NONE_ID || id != p)) ? 1 : 0;

    // wave scan 2: inclusive cumsum of new-segment flags (in-register)
    int sf = f;
#pragma unroll
    for (int off = 1; off < 32; off <<= 1) {
      int o = __shfl_up(sf, off, 32);
      if (lane >= off) sf += o;
    }
    if (lane == 31) s_wsum[wave] = sf;
    __syncthreads();                                   // B2

    int base = 0;
    for (int wi = 0; wi < wave; ++wi) base += s_wsum[wi];

    if (t < T) {
      if (valid) {
        const int sidx = carrySeg + base + sf - 1;
        seg[b * T + t] = sidx;
        if (f) seg_start[b * T + sidx] = t;
        atomicAdd(&counts[b * T + sidx], 1);
      } else {
        seg[b * T + t] = T;  // drop bucket
      }
    }

    // rowlim: wave max reduce, one LDS atomic per wave
    int tm = valid ? t + 1 : 0;
#pragma unroll
    for (int off = 16; off >= 1; off >>= 1) {
      int o = __shfl_xor(tm, off, 32);
      tm = (o > tm) ? o : tm;
    }
    if (lane == 0) atomicMax(&sh_row, tm);

    // chunk carry update
    if (tid == 0) {
      int nl = carryLast;
      for (int wi = 7; wi >= 0; --wi) {
        int x = s_wval[wi];
        if (x != NONE_ID) { nl = x; break; }
      }
      sh_last = nl;
      int tot = 0;
      for (int wi = 0; wi < 8; ++wi) tot += s_wsum[wi];
      sh_seg = carrySeg + tot;
    }
    __syncthreads();                                   // B3
  }
  if (tid == 0) { nseg_arr[b] = sh_seg; rowlim_arr[b] = sh_row; }
}

// ---------------------------------------------------------------------------
// Kernel 2: WMMA segment-mean + expand. One block per (sample, group of 16
// consecutive segments). 256 threads = 8 waves; each wave owns 4 column
// tiles of 16 (8 * 64 = 512 = D). Segment sums computed exactly as
// D += A(16x4 indicator, f32) x B(4x16 values, f32) with
// v_wmma_f32_16x16x4_f32 over row-chunks of 4.
// A/B VGPR layout (ISA 7.12.2, 32-bit 16x4): element r of the v2f holds
// K = r + (lane<16 ? 0 : 2); M/N = lane&15. C/D: VGPR r -> M = r + (lane<16?0:8).
//
// Loads are UNCONDITIONAL (row index clamped to T-1): rows in [rowEnd, T)
// already have seg outside [g0, g0+16) so the indicator is 0; for t >= T the
// indicator is forced to 0. values is finite everywhere -> 0*x == 0 exactly.
// All addressing is base-pointer + 32-bit index (per-sample offsets < 2^31).
// ---------------------------------------------------------------------------
__global__ void opap_mean_kernel(const float* __restrict__ values,
                                 const int* __restrict__ seg,
                                 const int* __restrict__ seg_start,
                                 const int* __restrict__ counts,
                                 const int* __restrict__ nseg_arr,
                                 const int* __restrict__ rowlim_arr,
                                 float* __restrict__ out,
                                 int T, int D) {
  const int b = blockIdx.y;
  const int g0 = blockIdx.x << 4;          // first segment of this group
  const int nseg = nseg_arr[b];
  if (g0 >= nseg) return;

  const int rowStart = seg_start[b * T + g0];
  const int rowEnd = (g0 + 16 < nseg) ? seg_start[b * T + g0 + 16]
                                      : rowlim_arr[b];

  const int tid = threadIdx.x;
  const int lane = tid & 31;
  const int wave = tid >> 5;
  const int m16 = lane & 15;
  const int kbase = (lane < 16) ? 0 : 2;   // K offset for this lane half
  const int colBase = wave * 64;

  const float* __restrict__ vptr = values + (size_t)b * T * D;  // sample base
  float* __restrict__ outB = out + (size_t)b * T * D;
  const int* __restrict__ segRow = seg + b * T;

  v8f acc0 = {}; v8f acc1 = {}; v8f acc2 = {}; v8f acc3 = {};

  const int tmax = T - 1;
  const int c0 = colBase + m16;

  for (int t0 = rowStart; t0 < rowEnd; t0 += 4) {
    const int tA = t0 + kbase;             // K = kbase   row
    const int tB = tA + 1;                 // K = kbase+1 row
    const int tAc = (tA < tmax) ? tA : tmax;
    const int tBc = (tB < tmax) ? tB : tmax;

    int sA = segRow[tAc]; if (tA >= T) sA = BIG_ID;
    int sB = segRow[tBc]; if (tB >= T) sB = BIG_ID;

    v2f a;                                 // 16x4 indicator matrix
    a.x = (sA - g0 == m16) ? 1.0f : 0.0f;
    a.y = (sB - g0 == m16) ? 1.0f : 0.0f;

    const int iA = tAc * D + c0;           // 32-bit indices -> saddr+voffset
    const int iB = tBc * D + c0;

    v2f b0, b1, b2, b3;                    // 4x16 value tiles
    b0.x = vptr[iA     ];  b0.y = vptr[iB     ];
    b1.x = vptr[iA + 16];  b1.y = vptr[iB + 16];
    b2.x = vptr[iA + 32];  b2.y = vptr[iB + 32];
    b3.x = vptr[iA + 48];  b3.y = vptr[iB + 48];

    acc0 = __builtin_amdgcn_wmma_f32_16x16x4_f32(false, a, false, b0, (short)0, acc0, false, false);
    acc1 = __builtin_amdgcn_wmma_f32_16x16x4_f32(false, a, false, b1, (short)0, acc1, false, false);
    acc2 = __builtin_amdgcn_wmma_f32_16x16x4_f32(false, a, false, b2, (short)0, acc2, false, false);
    acc3 = __builtin_amdgcn_wmma_f32_16x16x4_f32(false, a, false, b3, (short)0, acc3, false, false);
  }

  // means -> LDS (row stride 516 floats: M and M+8 land in different banks;
  // 516*4 bytes is a multiple of 16 so float4 reads stay aligned).
  // Exact IEEE division per element to match the reference's sums/counts.
  __shared__ float aggLocal[16][516];
  const int Mlo = (lane < 16) ? 0 : 8;
#pragma unroll
  for (int r = 0; r < 8; ++r) {
    const int M = r + Mlo;
    const int s = g0 + M;
    float cnt = 1.0f;
    if (s < nseg) {
      int ci = counts[b * T + s];
      cnt = (float)(ci < 1 ? 1 : ci);
    }
    aggLocal[M][c0     ] = acc0[r] / cnt;
    aggLocal[M][c0 + 16] = acc1[r] / cnt;
    aggLocal[M][c0 + 32] = acc2[r] / cnt;
    aggLocal[M][c0 + 48] = acc3[r] / cnt;
  }
  __syncthreads();

  // expand: broadcast each segment's mean row to all its phone positions.
  // Two rows per iteration, 128 threads x float4 (b128) per row.
  const int half = tid >> 7;               // 0 or 1: which row of the pair
  const int col4 = (tid & 127) << 2;       // float4 column offset
  for (int t = rowStart + half; t < rowEnd; t += 2) {
    const unsigned m = (unsigned)(segRow[t] - g0);
    if (m < 16u) {
      float4 v = *(const float4*)&aggLocal[m][col4];
      *(float4*)(outB + t * D + col4) = v;
    }
  }
}

// ---------------------------------------------------------------------------
// Kernel 3: zero drop-bucket rows (padding / mid-stream invalid).
// ---------------------------------------------------------------------------
__global__ void opap_zero_kernel(const int* __restrict__ seg,
                                 float* __restrict__ out, int T, int D) {
  const int b = blockIdx.y;
  const int t = blockIdx.x;
  if (seg[b * T + t] != T) return;
  float* outB = out + (size_t)b * T * D;
  float4 z = make_float4(0.f, 0.f, 0.f, 0.f);
  *(float4*)(outB + t * D + threadIdx.x * 4) = z;
}

// ---------------------------------------------------------------------------
extern "C" void kernel_launch(void* const* d_in, const int* in_sizes, int n_in,
                              void* d_out, int out_size, void* d_ws, size_t ws_size,
                              hipStream_t stream) {
  const int* word_ids = (const int*)d_in[0];
  const float* values = (const float*)d_in[1];
  float* out = (float*)d_out;

  const int T = 4096;
  const int B = in_sizes[0] / T;
  const int D = in_sizes[1] / in_sizes[0];   // 512

  char* ws = (char*)d_ws;
  int* seg       = (int*)ws; ws += (size_t)B * T * sizeof(int);
  int* seg_start = (int*)ws; ws += (size_t)B * T * sizeof(int);
  int* counts    = (int*)ws; ws += (size_t)B * T * sizeof(int);
  int* nseg      = (int*)ws; ws += (size_t)B * sizeof(int);
  int* rowlim    = (int*)ws;

  hipMemsetAsync(counts, 0, (size_t)B * T * sizeof(int), stream);

  opap_scan_kernel<<<B, 256, 0, stream>>>(word_ids, seg, seg_start, counts,
                                          nseg, rowlim, T);

  dim3 gridMean((T + 15) / 16, B);
  opap_mean_kernel<<<gridMean, 256, 0, stream>>>(values, seg, seg_start, counts,
                                                 nseg, rowlim, out, T, D);

  dim3 gridZero(T, B);
  opap_zero_kernel<<<gridZero, D / 4, 0, stream>>>(seg, out, T, D);
}